// ReLuPCA_30554397343924
// MI455X (gfx1250) — compile-verified
//
#include <hip/hip_runtime.h>
#include <stdint.h>

#define BATCH 32
#define HW    3136               // 56*56 (multiple of 16 -> tiles never cross batch)
#define CH    256
#define NPIX  (BATCH * HW)       // 100352
#define WAVES 8
#define TPB   (WAVES * 32)       // 256 threads, 8 wave32
#define UQ    1044               // per column-quad stride (floats); UQ%64==20 -> no bank conflicts
#define STATS_BYTES 8192
#define PROJ_FLOATS ((size_t)NPIX * CH)

typedef __attribute__((ext_vector_type(2))) float v2f;
typedef __attribute__((ext_vector_type(8))) float v8f;

// D(16x16,f32) = A(16x4,f32) x B(4x16,f32) + C   -- v_wmma_f32_16x16x4_f32
__device__ __forceinline__ v8f wmma4(v2f a, v2f b, v8f c) {
  return __builtin_amdgcn_wmma_f32_16x16x4_f32(false, a, false, b, (short)0, c, false, false);
}

// order-preserving float <-> uint encoding (for atomic min/max on negatives)
__device__ __forceinline__ unsigned f2s(float f) {
  unsigned u = __float_as_uint(f);
  return (u & 0x80000000u) ? ~u : (u | 0x80000000u);
}
__device__ __forceinline__ float s2f(unsigned s) {
  unsigned u = (s & 0x80000000u) ? (s & 0x7FFFFFFFu) : ~s;
  return __uint_as_float(u);
}

// async memory->LDS preload of u in column-quad interleaved layout:
//   LDS_word(k,c) = (c>>2)*UQ + 4k + (c&3)
__device__ __forceinline__ void load_u_async(const float* __restrict__ u,
                                             float* u_l, int tid) {
  unsigned base = (unsigned)(uintptr_t)u_l;     // LDS aperture: low 32 bits = LDS offset
#pragma unroll 4
  for (int i = tid; i < CH * CH / 4; i += TPB) {
    int k = i >> 6, cq = i & 63;                // 64 column-quads
    unsigned dst = base + ((unsigned)(cq * UQ + 4 * k) << 2);
    unsigned long long src = (unsigned long long)(uintptr_t)(u + (size_t)k * CH + 4 * cq);
    asm volatile("global_load_async_to_lds_b128 %0, %1, off"
                 :: "v"(dst), "v"(src) : "memory");
  }
  asm volatile("s_wait_asynccnt 0x0" ::: "memory");
}

// GEMM1 over k: XPOSE -> D = relu(x)^T x u-tile (pixels x channels, pass 2)
//               else  -> D = u.T-tile x relu(x) (channels x pixels, fallback pass 3)
template <bool XPOSE>
__device__ __forceinline__ void gemm1(const float* __restrict__ xb,
                                      const float* __restrict__ ub,
                                      int kk, v8f (&acc)[16]) {
  auto LOADK = [&](int k0, v2f& im, v2f (&ut)[16]) {
    int ka = k0 + kk;
    im.x = fmaxf(xb[(size_t)ka * HW], 0.f);
    im.y = fmaxf(xb[(size_t)(ka + 1) * HW], 0.f);
#pragma unroll
    for (int t = 0; t < 16; t++) {
      ut[t].x = ub[(size_t)t * 4 * UQ + 4 * ka];
      ut[t].y = ub[(size_t)t * 4 * UQ + 4 * ka + 4];
    }
  };
  v2f imA, imB, utA[16], utB[16];
  LOADK(0, imA, utA);
#pragma unroll 1
  for (int k0 = 0; k0 < CH; k0 += 8) {          // 2x unrolled, double buffered
    LOADK(k0 + 4, imB, utB);
#pragma unroll
    for (int t = 0; t < 16; t++) {
      if (XPOSE) acc[t] = wmma4(imA, utA[t], acc[t]);
      else       acc[t] = wmma4(utA[t], imA, acc[t]);
    }
    if (k0 + 8 < CH) LOADK(k0 + 8, imA, utA);
#pragma unroll
    for (int t = 0; t < 16; t++) {
      if (XPOSE) acc[t] = wmma4(imB, utB[t], acc[t]);
      else       acc[t] = wmma4(utB[t], imB, acc[t]);
    }
  }
}

// GEMM2: rec = u @ q from prebuilt B-layout groups bq[kt][g]; adds mean, stores.
// kt loop MUST be fully unrolled: constant indices keep bq in VGPRs (no scratch).
__device__ __forceinline__ void gemm2_out(const float* __restrict__ u_l,
                                          const v2f (&bq)[16][4],
                                          const float* __restrict__ mn_l,
                                          float* __restrict__ outp,
                                          int lm, int hf, int kk) {
  auto LOADA2 = [&](int kt, int oc, v2f (&a2)[4]) {
#pragma unroll
    for (int g = 0; g < 4; g++)
      a2[g] = *(const v2f*)(u_l + (size_t)(kt * 4 + g) * UQ + 64 * oc + 4 * lm + kk);
  };
#pragma unroll 1
  for (int oh = 0; oh < 2; oh++) {
    v8f r[8] = {};
#pragma unroll
    for (int kt = 0; kt < 16; kt++) {
      v2f a2A[4], a2B[4];
      LOADA2(kt, oh * 8, a2A);
#pragma unroll
      for (int o = 0; o < 8; o += 2) {
        LOADA2(kt, oh * 8 + o + 1, a2B);
#pragma unroll
        for (int g = 0; g < 4; g++) r[o] = wmma4(a2A[g], bq[kt][g], r[o]);
        if (o + 2 < 8) LOADA2(kt, oh * 8 + o + 2, a2A);
#pragma unroll
        for (int g = 0; g < 4; g++) r[o + 1] = wmma4(a2B[g], bq[kt][g], r[o + 1]);
      }
    }
#pragma unroll
    for (int o = 0; o < 8; o++) {
#pragma unroll
      for (int v = 0; v < 8; v++) {
        int c = (oh * 8 + o) * 16 + hf * 8 + v;
        outp[(size_t)c * HW] = r[o][v] + mn_l[c];
      }
    }
  }
}

// ---------------- small kernels ----------------
__global__ void k_init(float* wsum, unsigned* wmin, unsigned* wmax) {
  int c = threadIdx.x;
  wsum[c] = 0.0f; wmin[c] = 0xFFFFFFFFu; wmax[c] = 0u;
}

__global__ __launch_bounds__(256) void k_mean(const float* __restrict__ x,
                                              float* __restrict__ wsum) {
  __shared__ float red[256];
  int bc = blockIdx.x;
  const float4* row = (const float4*)(x + (size_t)bc * HW);
  float s = 0.f;
  for (int i = threadIdx.x; i < HW / 4; i += 256) {
    float4 v = row[i];
    s += fmaxf(v.x, 0.f) + fmaxf(v.y, 0.f) + fmaxf(v.z, 0.f) + fmaxf(v.w, 0.f);
  }
  red[threadIdx.x] = s;
  __syncthreads();
  for (int o = 128; o > 0; o >>= 1) {
    if (threadIdx.x < o) red[threadIdx.x] += red[threadIdx.x + o];
    __syncthreads();
  }
  if (threadIdx.x == 0) atomicAdd(&wsum[bc & (CH - 1)], red[0]);
}

__global__ void k_mean_fin(const float* __restrict__ u, const float* wsum,
                           float* wmean, float* wpmn) {
  __shared__ float mn_s[CH];
  int c = threadIdx.x;
  float m = wsum[c] * (1.0f / (float)NPIX);
  wmean[c] = m; mn_s[c] = m;
  __syncthreads();
  float s = 0.f;
  for (int k = 0; k < CH; k++) s += u[(size_t)k * CH + c] * mn_s[k];
  wpmn[c] = s;
}

__global__ void k_scale_fin(const unsigned* wmin, const unsigned* wmax,
                            float* wscale, float* winv, float* wdmin) {
  int c = threadIdx.x;
  float dmin = s2f(wmin[c]), dmax = s2f(wmax[c]);
  float rng = dmax - dmin;
  wscale[c] = 255.0f / rng;
  winv[c]   = rng * (1.0f / 255.0f);
  wdmin[c]  = dmin;
}

// ---------------- pass 2: projection + clamp + min/max (+ optional proj store) ----
template <bool STORE>
__global__ __launch_bounds__(TPB, 1) void k_proj_minmax(
    const float* __restrict__ x, const float* __restrict__ u,
    const float* __restrict__ clampv, const float* __restrict__ wpmn,
    unsigned* __restrict__ gmin, unsigned* __restrict__ gmax,
    float* __restrict__ proj) {
  extern __shared__ float sm[];
  float* u_l  = sm;
  float* pm_l = u_l + 64 * UQ;
  float* cv_l = pm_l + CH;
  unsigned* bmin = (unsigned*)(cv_l + CH);
  unsigned* bmax = bmin + CH;

  int tid = threadIdx.x;
  load_u_async(u, u_l, tid);
  if (tid < CH) {
    pm_l[tid] = wpmn[tid]; cv_l[tid] = clampv[tid];
    bmin[tid] = 0xFFFFFFFFu; bmax[tid] = 0u;
  }
  __syncthreads();

  int lane = tid & 31, hf = lane >> 4, lm = lane & 15, kk = hf * 2;
  int T = blockIdx.x * WAVES + (tid >> 5);      // 16-pixel tile id
  int P = T * 16, b = P / HW, hw = P % HW;
  const float* xb = x + (size_t)b * CH * HW + hw + lm;
  const float* ub = u_l + (lm >> 2) * UQ + (lm & 3);

  v8f acc[16] = {};
  gemm1<true>(xb, ub, kk, acc);

  // tile store base: word(cp,n) = (cp>>1)*32 + 2n + (cp&1)
  float* pp = STORE ? proj + (size_t)T * 4096 + (lm >> 1) * 32 + (lm & 1) + hf * 16 : nullptr;

#pragma unroll
  for (int t = 0; t < 16; t++) {
    int cp = t * 16 + lm;
    float cv = cv_l[cp];
    float pm = pm_l[cp];
    float mnv = cv, mxv = -cv;
#pragma unroll
    for (int v = 0; v < 8; v++) {
      float val = fminf(fmaxf(acc[t][v] - pm, -cv), cv);
      if (STORE) pp[t * 256 + v * 2] = val;
      mnv = fminf(mnv, val);
      mxv = fmaxf(mxv, val);
    }
    mnv = fminf(mnv, __shfl_xor(mnv, 16, 32));
    mxv = fmaxf(mxv, __shfl_xor(mxv, 16, 32));
    if (hf == 0) {
      atomicMin(&bmin[cp], f2s(mnv));
      atomicMax(&bmax[cp], f2s(mxv));
    }
  }
  __syncthreads();
  if (tid < CH) {
    atomicMin(&gmin[tid], bmin[tid]);
    atomicMax(&gmax[tid], bmax[tid]);
  }
}

// ---------------- pass 3a: quantize stored proj + GEMM2 (no GEMM1 recompute) ----
__global__ __launch_bounds__(TPB, 1) void k_rec_proj(
    const float* __restrict__ proj, const float* __restrict__ u,
    const float* __restrict__ wmean, const float* __restrict__ wscale,
    const float* __restrict__ winv, const float* __restrict__ wdmin,
    float* __restrict__ out) {
  extern __shared__ float sm[];
  float* u_l  = sm;
  float* sc_l = u_l + 64 * UQ;
  float* iv_l = sc_l + CH;
  float* dm_l = iv_l + CH;
  float* mn_l = dm_l + CH;

  int tid = threadIdx.x;
  load_u_async(u, u_l, tid);
  if (tid < CH) {
    sc_l[tid] = wscale[tid]; iv_l[tid] = winv[tid];
    dm_l[tid] = wdmin[tid];  mn_l[tid] = wmean[tid];
  }
  __syncthreads();

  int lane = tid & 31, hf = lane >> 4, lm = lane & 15, kk = hf * 2;
  int T = blockIdx.x * WAVES + (tid >> 5);
  int P = T * 16, b = P / HW, hw = P % HW;

  // coalesced b64 loads: channel pair (cpA,cpA+1) at column lm, already B-layout
  const float* lp = proj + (size_t)T * 4096 + lm * 2 + hf * 32;
  v2f bq[16][4];
#pragma unroll
  for (int kt = 0; kt < 16; kt++) {
#pragma unroll
    for (int g = 0; g < 4; g++) {
      v2f pr = *(const v2f*)(lp + kt * 256 + g * 64);
      int ca = kt * 16 + 4 * g + 2 * hf;        // lane-uniform per half-wave
      float d0 = dm_l[ca], d1 = dm_l[ca + 1];
      pr.x = rintf((pr.x - d0) * sc_l[ca])     * iv_l[ca]     + d0;
      pr.y = rintf((pr.y - d1) * sc_l[ca + 1]) * iv_l[ca + 1] + d1;
      bq[kt][g] = pr;
    }
  }

  float* outp = out + (size_t)b * CH * HW + hw + lm;
  gemm2_out(u_l, bq, mn_l, outp, lm, hf, kk);
}

// ---------------- pass 3b (fallback): recompute GEMM1, quantize, GEMM2 ----------
__global__ __launch_bounds__(TPB, 1) void k_rec_full(
    const float* __restrict__ x, const float* __restrict__ u,
    const float* __restrict__ clampv, const float* __restrict__ wmean,
    const float* __restrict__ wpmn, const float* __restrict__ wscale,
    const float* __restrict__ winv, const float* __restrict__ wdmin,
    float* __restrict__ out) {
  extern __shared__ float sm[];
  float* u_l  = sm;
  float* pm_l = u_l + 64 * UQ;
  float* cv_l = pm_l + CH;
  float* sc_l = cv_l + CH;
  float* iv_l = sc_l + CH;
  float* dm_l = iv_l + CH;
  float* mn_l = dm_l + CH;

  int tid = threadIdx.x;
  load_u_async(u, u_l, tid);
  if (tid < CH) {
    pm_l[tid] = wpmn[tid]; cv_l[tid] = clampv[tid];
    sc_l[tid] = wscale[tid]; iv_l[tid] = winv[tid];
    dm_l[tid] = wdmin[tid]; mn_l[tid] = wmean[tid];
  }
  __syncthreads();

  int lane = tid & 31, hf = lane >> 4, lm = lane & 15, kk = hf * 2;
  int T = blockIdx.x * WAVES + (tid >> 5);
  int P = T * 16, b = P / HW, hw = P % HW;
  const float* xb = x + (size_t)b * CH * HW + hw + lm;
  const float* ub = u_l + (lm >> 2) * UQ + (lm & 3);

  v8f q[16] = {};
  gemm1<false>(xb, ub, kk, q);

#pragma unroll
  for (int t = 0; t < 16; t++) {
#pragma unroll
    for (int v = 0; v < 8; v++) {
      int cp = t * 16 + hf * 8 + v;
      float cv = cv_l[cp];
      float val = fminf(fmaxf(q[t][v] - pm_l[cp], -cv), cv);
      float dm = dm_l[cp];
      q[t][v] = rintf((val - dm) * sc_l[cp]) * iv_l[cp] + dm;
    }
  }

  // convert q (C/D layout) into B-layout groups via half-wave swaps
  v2f bq[16][4];
#pragma unroll
  for (int kt = 0; kt < 16; kt++) {
#pragma unroll
    for (int g = 0; g < 2; g++) {
      float s0 = __shfl_xor(q[kt][4 * g + 2], 16, 32);
      float s1 = __shfl_xor(q[kt][4 * g + 3], 16, 32);
      bq[kt][g].x = hf ? s0 : q[kt][4 * g];
      bq[kt][g].y = hf ? s1 : q[kt][4 * g + 1];
    }
#pragma unroll
    for (int g = 2; g < 4; g++) {
      float s0 = __shfl_xor(q[kt][4 * g - 8], 16, 32);
      float s1 = __shfl_xor(q[kt][4 * g - 7], 16, 32);
      bq[kt][g].x = hf ? q[kt][4 * g - 6] : s0;
      bq[kt][g].y = hf ? q[kt][4 * g - 5] : s1;
    }
  }

  float* outp = out + (size_t)b * CH * HW + hw + lm;
  gemm2_out(u_l, bq, mn_l, outp, lm, hf, kk);
}

extern "C" void kernel_launch(void* const* d_in, const int* in_sizes, int n_in,
                              void* d_out, int out_size, void* d_ws, size_t ws_size,
                              hipStream_t stream) {
  (void)in_sizes; (void)n_in; (void)out_size;
  const float* x  = (const float*)d_in[0];
  const float* u  = (const float*)d_in[1];
  const float* cv = (const float*)d_in[2];
  float* out = (float*)d_out;

  float*    wsum  = (float*)d_ws;
  float*    wmean = wsum + CH;
  float*    wpmn  = wmean + CH;
  unsigned* wmin  = (unsigned*)(wpmn + CH);
  unsigned* wmax  = wmin + CH;
  float*    wscale = (float*)(wmax + CH);
  float*    winv   = wscale + CH;
  float*    wdmin  = winv + CH;
  float*    proj   = (float*)((char*)d_ws + STATS_BYTES);

  const bool use_proj = ws_size >= STATS_BYTES + PROJ_FLOATS * 4;  // deterministic
  const int nblk = NPIX / (WAVES * 16);                            // 784, exact
  const size_t shm2  = (size_t)(64 * UQ + 2 * CH) * 4 + (size_t)(2 * CH) * 4;
  const size_t shm3a = (size_t)(64 * UQ + 4 * CH) * 4;
  const size_t shm3b = (size_t)(64 * UQ + 6 * CH) * 4;

  k_init<<<1, 256, 0, stream>>>(wsum, wmin, wmax);
  k_mean<<<BATCH * CH, 256, 0, stream>>>(x, wsum);
  k_mean_fin<<<1, 256, 0, stream>>>(u, wsum, wmean, wpmn);
  if (use_proj) {
    k_proj_minmax<true><<<nblk, TPB, shm2, stream>>>(x, u, cv, wpmn, wmin, wmax, proj);
    k_scale_fin<<<1, 256, 0, stream>>>(wmin, wmax, wscale, winv, wdmin);
    k_rec_proj<<<nblk, TPB, shm3a, stream>>>(proj, u, wmean, wscale, winv, wdmin, out);
  } else {
    k_proj_minmax<false><<<nblk, TPB, shm2, stream>>>(x, u, cv, wpmn, wmin, wmax, proj);
    k_scale_fin<<<1, 256, 0, stream>>>(wmin, wmax, wscale, winv, wdmin);
    k_rec_full<<<nblk, TPB, shm3b, stream>>>(x, u, cv, wmean, wpmn, wscale, winv, wdmin, out);
  }
}